// Net_77584289235462
// MI455X (gfx1250) — compile-verified
//
#include <hip/hip_runtime.h>
#include <hip/hip_bf16.h>
#include <math.h>

typedef __attribute__((ext_vector_type(2))) float v2f;
typedef __attribute__((ext_vector_type(8))) float v8f;

#define NNODES 100000        // multiple of 16 -> 6250 row tiles, no row guards
#define NEDGES 3200000
#define NGRAPH 512
#define BN_EPS 1e-5f
#define ROWTILES (NNODES / 16)

// ---------- small utility kernels ----------
__global__ void k_fill_f32(float* __restrict__ p, float v, int n) {
    int i = blockIdx.x * blockDim.x + threadIdx.x;
    if (i < n) p[i] = v;
}
__global__ void k_fill_u32(unsigned* __restrict__ p, unsigned v, int n) {
    int i = blockIdx.x * blockDim.x + threadIdx.x;
    if (i < n) p[i] = v;
}

// deg[dst] += 1 per edge (self loops pre-seeded by filling 1.0)
__global__ void k_degree(const long long* __restrict__ dst, float* __restrict__ deg, int ne) {
    int i = blockIdx.x * blockDim.x + threadIdx.x;
    if (i < ne) atomicAdd(&deg[(int)dst[i]], 1.0f);
}
__global__ void k_rsqrt_inplace(float* __restrict__ d, int n) {
    int i = blockIdx.x * blockDim.x + threadIdx.x;
    if (i < n) d[i] = rsqrtf(d[i]);   // deg >= 1 always (self loops)
}

// ---------- WMMA GEMM: S[i][f] = (H[i][:] . W[:][f]) * dis[i] ----------
// One wave computes a 16 x (16*CPT) output tile via V_WMMA_F32_16X16X4_F32.
// A layout: lanes 0-15 -> (k0,k0+1), lanes 16-31 -> (k0+2,k0+3), row = row0 + lane%16
// B layout (symmetric): same k split, col = col0 + lane%16
// C/D: VGPR v -> row row0 + v + 8*(lane>=16), col col0 + lane%16
template<int K, int F, int CPT>
__global__ __launch_bounds__(128)
void k_gemm_scale(const float* __restrict__ H, const float* __restrict__ W,
                  const float* __restrict__ dis, float* __restrict__ S) {
    const int wave = threadIdx.x >> 5;
    const int lane = threadIdx.x & 31;
    const int rowTile = blockIdx.x * 4 + wave;
    if (rowTile >= ROWTILES) return;            // wave-uniform; EXEC stays all-ones
    const int row0 = rowTile * 16;
    const int half = lane >> 4;                  // 0 or 1
    const int l16  = lane & 15;
    const int arow = row0 + l16;
    const int colBase = blockIdx.y * (16 * CPT);

    v8f c[CPT] = {};

#pragma unroll
    for (int kk = 0; kk < K; kk += 4) {
        const int ka = kk + half * 2;            // this lane's (k, k+1)
        v2f a;
        if constexpr (K >= 4) {
            a = *(const v2f*)(H + (size_t)arow * K + ka);   // global_load_b64
        } else {
            a.x = (ka     < K) ? H[(size_t)arow * K + ka]     : 0.f;
            a.y = (ka + 1 < K) ? H[(size_t)arow * K + ka + 1] : 0.f;
        }
#pragma unroll
        for (int t = 0; t < CPT; ++t) {
            const int bcol = colBase + t * 16 + l16;
            const bool colOK = (F % 16 == 0) || (bcol < F);  // constant-true unless F==8
            v2f b; b.x = 0.f; b.y = 0.f;
            if constexpr (K >= 4) {
                if (colOK) {
                    b.x = W[(size_t)ka       * F + bcol];
                    b.y = W[(size_t)(ka + 1) * F + bcol];
                }
            } else {
                if (colOK && ka     < K) b.x = W[(size_t)ka       * F + bcol];
                if (colOK && ka + 1 < K) b.y = W[(size_t)(ka + 1) * F + bcol];
            }
            c[t] = __builtin_amdgcn_wmma_f32_16x16x4_f32(false, a, false, b,
                                                         (short)0, c[t], false, false);
        }
    }

#pragma unroll
    for (int v = 0; v < 8; ++v) {
        const int r = row0 + v + 8 * half;
        const float dr = dis[r];
#pragma unroll
        for (int t = 0; t < CPT; ++t) {
            const int bcol = colBase + t * 16 + l16;
            if ((F % 16 == 0) || (bcol < F))
                S[(size_t)r * F + bcol] = c[t][v] * dr;
        }
    }
}

// ---------- edge scatter: acc[dst][4q..] += S[src][4q..] (float4 per lane) ----------
template<int F>
__global__ __launch_bounds__(256)
void k_scatter(const long long* __restrict__ src, const long long* __restrict__ dst,
               const float* __restrict__ S, float* __restrict__ acc) {
    constexpr int QPE = F >> 2;                  // float4 chunks per edge
    const long t = blockIdx.x * (long)blockDim.x + threadIdx.x;
    if (t >= (long)NEDGES * QPE) return;
    const int e = (int)(t / QPE);                // / and % by constant -> shifts
    const int q = (int)(t % QPE);
    const int s = (int)src[e];
    const int d = (int)dst[e];
    const float4 val = *(const float4*)(S + (size_t)s * F + q * 4);  // global_load_b128
    float* ap = acc + (size_t)d * F + q * 4;
    atomicAdd(ap + 0, val.x);
    atomicAdd(ap + 1, val.y);
    atomicAdd(ap + 2, val.z);
    atomicAdd(ap + 3, val.w);
}

// ---------- epilogue: conv = (acc + s_self)*dis + b ; relu ; BN partial sums ----------
template<int F>
__global__ __launch_bounds__(256)
void k_epilogue(const float* __restrict__ acc, const float* __restrict__ S,
                const float* __restrict__ dis, const float* __restrict__ bias,
                float* __restrict__ Hout, float* __restrict__ bnsum,
                float* __restrict__ bnsq) {
    __shared__ float ssum[F];
    __shared__ float ssq[F];
    for (int f = threadIdx.x; f < F; f += blockDim.x) { ssum[f] = 0.f; ssq[f] = 0.f; }
    __syncthreads();

    const long idx = blockIdx.x * (long)blockDim.x + threadIdx.x;
    if (idx < (long)NNODES * F) {
        const int i = (int)(idx / F);
        const int f = (int)(idx % F);
        const float v = (acc[idx] + S[idx]) * dis[i] + bias[f];
        const float r = fmaxf(v, 0.f);
        Hout[idx] = r;
        atomicAdd(&ssum[f], r);                  // ds_add_f32
        atomicAdd(&ssq[f], r * r);
    }
    __syncthreads();
    for (int f = threadIdx.x; f < F; f += blockDim.x) {
        if (ssum[f] != 0.f) atomicAdd(&bnsum[f], ssum[f]);
        if (ssq[f]  != 0.f) atomicAdd(&bnsq[f],  ssq[f]);
    }
}

__global__ void k_bn_params(const float* __restrict__ bnsum, const float* __restrict__ bnsq,
                            const float* __restrict__ g, const float* __restrict__ be,
                            float* __restrict__ aOut, float* __restrict__ cOut, int F) {
    const int f = blockIdx.x * blockDim.x + threadIdx.x;
    if (f >= F) return;
    const float m   = bnsum[f] / (float)NNODES;
    const float var = fmaxf(bnsq[f] / (float)NNODES - m * m, 0.f);
    const float a   = g[f] * rsqrtf(var + BN_EPS);
    aOut[f] = a;
    cOut[f] = be[f] - m * a;
}

template<int F>
__global__ __launch_bounds__(256)
void k_bn_apply(float* __restrict__ H, const float* __restrict__ a,
                const float* __restrict__ c) {
    const long idx = blockIdx.x * (long)blockDim.x + threadIdx.x;
    if (idx < (long)NNODES * F) {
        const int f = (int)(idx % F);
        H[idx] = H[idx] * a[f] + c[f];
    }
}

// ---------- segment max pooling via order-preserving uint encoding ----------
__device__ __forceinline__ unsigned enc_f32(float x) {
    unsigned u = __float_as_uint(x);
    return u ^ (unsigned)(((int)u >> 31) | 0x80000000);
}
__device__ __forceinline__ float dec_f32(unsigned e) {
    unsigned u = (e & 0x80000000u) ? (e ^ 0x80000000u) : ~e;
    return __uint_as_float(u);
}

__global__ __launch_bounds__(256)
void k_pool(const float* __restrict__ H, const long long* __restrict__ batch,
            unsigned* __restrict__ pooled) {
    const long idx = blockIdx.x * (long)blockDim.x + threadIdx.x;
    if (idx >= (long)NNODES * 128) return;
    const int i = (int)(idx >> 7);
    const int f = (int)(idx & 127);
    const int g = (int)batch[i];
    atomicMax(&pooled[g * 128 + f], enc_f32(H[idx]));  // global_atomic_max_u32
}

// ---------- head: logits = pooled @ Wl + bl ; log_softmax ----------
__global__ void k_head(const unsigned* __restrict__ pooled, const float* __restrict__ Wl,
                       const float* __restrict__ bl, float* __restrict__ out) {
    const int g = blockIdx.x * blockDim.x + threadIdx.x;
    if (g >= NGRAPH) return;
    float l[3];
    for (int k = 0; k < 3; ++k) {
        float acc = bl[k];
        for (int f = 0; f < 128; ++f)
            acc += dec_f32(pooled[g * 128 + f]) * Wl[f * 3 + k];
        l[k] = acc;
    }
    const float m = fmaxf(l[0], fmaxf(l[1], l[2]));
    const float s = expf(l[0] - m) + expf(l[1] - m) + expf(l[2] - m);
    const float lse = m + logf(s);
    out[g * 3 + 0] = l[0] - lse;
    out[g * 3 + 1] = l[1] - lse;
    out[g * 3 + 2] = l[2] - lse;
}

// ---------- per-layer driver (host) ----------
struct LayerBufs {
    const long long* srcI; const long long* dstI;
    float* dis; float* S; float* acc; float* Hbuf;
    float* bnsum; float* bnsq; float* bnA; float* bnC;
};

template<int K, int F, int CPT>
static void run_layer(const float* Hin, const float* W, const float* b,
                      const float* g, const float* be, const LayerBufs& B,
                      hipStream_t stream) {
    hipMemsetAsync(B.acc,   0, (size_t)NNODES * F * 4, stream);
    hipMemsetAsync(B.bnsum, 0, (size_t)F * 4, stream);
    hipMemsetAsync(B.bnsq,  0, (size_t)F * 4, stream);

    dim3 gg((ROWTILES + 3) / 4, (F + 16 * CPT - 1) / (16 * CPT));
    k_gemm_scale<K, F, CPT><<<gg, 128, 0, stream>>>(Hin, W, B.dis, B.S);

    const long tasks = (long)NEDGES * (F / 4);
    k_scatter<F><<<(int)((tasks + 255) / 256), 256, 0, stream>>>(B.srcI, B.dstI, B.S, B.acc);

    const long total = (long)NNODES * F;
    k_epilogue<F><<<(int)((total + 255) / 256), 256, 0, stream>>>(B.acc, B.S, B.dis, b,
                                                                  B.Hbuf, B.bnsum, B.bnsq);
    k_bn_params<<<1, 128, 0, stream>>>(B.bnsum, B.bnsq, g, be, B.bnA, B.bnC, F);
    k_bn_apply<F><<<(int)((total + 255) / 256), 256, 0, stream>>>(B.Hbuf, B.bnA, B.bnC);
}

// =====================================================================
extern "C" void kernel_launch(void* const* d_in, const int* in_sizes, int n_in,
                              void* d_out, int out_size, void* d_ws, size_t ws_size,
                              hipStream_t stream) {
    const float*     x   = (const float*)d_in[0];
    const long long* ei  = (const long long*)d_in[1];     // [2, E] int64
    const long long* bat = (const long long*)d_in[2];     // [N] int64
    const float* W1 = (const float*)d_in[3];  const float* b1 = (const float*)d_in[4];
    const float* g1 = (const float*)d_in[5];  const float* be1 = (const float*)d_in[6];
    const float* W2 = (const float*)d_in[7];  const float* b2 = (const float*)d_in[8];
    const float* g2 = (const float*)d_in[9];  const float* be2 = (const float*)d_in[10];
    const float* W3 = (const float*)d_in[11]; const float* b3 = (const float*)d_in[12];
    const float* g3 = (const float*)d_in[13]; const float* be3 = (const float*)d_in[14];
    const float* Wl = (const float*)d_in[15]; const float* bl = (const float*)d_in[16];

    // ---- workspace layout (256B aligned slices) ----
    char* ws = (char*)d_ws;
    size_t off = 0;
    auto carve = [&](size_t bytes) -> char* {
        char* p = ws + off;
        off += (bytes + 255) & ~(size_t)255;
        return p;
    };
    LayerBufs B;
    B.srcI = ei;                 // edge_index[0]
    B.dstI = ei + NEDGES;        // edge_index[1]
    B.dis   = (float*)carve((size_t)NNODES * 4);
    B.S     = (float*)carve((size_t)NNODES * 128 * 4);
    B.acc   = (float*)carve((size_t)NNODES * 128 * 4);
    B.Hbuf  = (float*)carve((size_t)NNODES * 128 * 4);
    B.bnsum = (float*)carve(128 * 4);
    B.bnsq  = (float*)carve(128 * 4);
    B.bnA   = (float*)carve(128 * 4);
    B.bnC   = (float*)carve(128 * 4);
    unsigned* pooled = (unsigned*)carve((size_t)NGRAPH * 128 * 4);
    (void)ws_size; (void)n_in; (void)in_sizes; (void)out_size;

    // ---- degree / symmetric norm ----
    k_fill_f32<<<(NNODES + 255) / 256, 256, 0, stream>>>(B.dis, 1.0f, NNODES); // self loops
    k_degree<<<(NEDGES + 255) / 256, 256, 0, stream>>>(B.dstI, B.dis, NEDGES);
    k_rsqrt_inplace<<<(NNODES + 255) / 256, 256, 0, stream>>>(B.dis, NNODES);

    // ---- three GCN layers ----
    run_layer<2,   8, 1>(x,      W1, b1, g1, be1, B, stream);
    run_layer<8,  32, 2>(B.Hbuf, W2, b2, g2, be2, B, stream);
    run_layer<32, 128, 2>(B.Hbuf, W3, b3, g3, be3, B, stream);

    // ---- pooling + head ----
    k_fill_u32<<<(NGRAPH * 128 + 255) / 256, 256, 0, stream>>>(pooled, 0x007FFFFFu,
                                                               NGRAPH * 128); // enc(-inf)
    const long ptot = (long)NNODES * 128;
    k_pool<<<(int)((ptot + 255) / 256), 256, 0, stream>>>(B.Hbuf, bat, pooled);
    k_head<<<(NGRAPH + 63) / 64, 64, 0, stream>>>(pooled, Wl, bl, (float*)d_out);
}